// SparsemaxBisectLoss_12421045420953
// MI455X (gfx1250) — compile-verified
//
#include <hip/hip_runtime.h>
#include <hip/hip_bf16.h>

#ifndef N_ITER
#define N_ITER 50
#endif
#define BLOCK 1024
#define MAX_D 32000   // 32000 floats = 128000 B of LDS row buffer
#define CAP   4096    // compacted-candidate capacity (16 KB LDS)
#define REG   8       // wave-0 register path: 32 lanes * REG = 256 candidates

typedef unsigned int u32x4 __attribute__((ext_vector_type(4)));
typedef int          i32x4 __attribute__((ext_vector_type(4)));
typedef int          i32x8 __attribute__((ext_vector_type(8)));

// ---------- wave32 primitives ----------
__device__ __forceinline__ float warp_reduce_sum(float v) {
#pragma unroll
  for (int o = 16; o > 0; o >>= 1) v += __shfl_xor(v, o, 32);
  return v;
}
__device__ __forceinline__ float warp_reduce_max(float v) {
#pragma unroll
  for (int o = 16; o > 0; o >>= 1) v = fmaxf(v, __shfl_xor(v, o, 32));
  return v;
}

__device__ __forceinline__ float block_reduce_sum(float v, float* scr) {
  const int tid = threadIdx.x, lane = tid & 31, wid = tid >> 5;
  const int nw = blockDim.x >> 5;
  v = warp_reduce_sum(v);
  if (lane == 0) scr[wid] = v;
  __syncthreads();
  if (wid == 0) {
    float t = (lane < nw) ? scr[lane] : 0.0f;
    t = warp_reduce_sum(t);
    if (lane == 0) scr[0] = t;
  }
  __syncthreads();
  float r = scr[0];
  __syncthreads();
  return r;
}

__device__ __forceinline__ float block_reduce_max(float v, float* scr) {
  const int tid = threadIdx.x, lane = tid & 31, wid = tid >> 5;
  const int nw = blockDim.x >> 5;
  v = warp_reduce_max(v);
  if (lane == 0) scr[wid] = v;
  __syncthreads();
  if (wid == 0) {
    float t = (lane < nw) ? scr[lane] : -3.402823466e38f;
    t = warp_reduce_max(t);
    if (lane == 0) scr[0] = t;
  }
  __syncthreads();
  float r = scr[0];
  __syncthreads();
  return r;
}

// deterministic block-wide exclusive prefix scan (1024 threads, 32 waves)
__device__ __forceinline__ int block_exscan_total(int v, int* iscr, int* total_out) {
  const int tid = threadIdx.x, lane = tid & 31, wid = tid >> 5;
  const int nw = blockDim.x >> 5;
  int x = v;
#pragma unroll
  for (int o = 1; o < 32; o <<= 1) {
    int t = __shfl_up(x, o, 32);
    if (lane >= o) x += t;
  }
  if (lane == 31) iscr[wid] = x;  // wave totals
  __syncthreads();
  if (wid == 0) {
    int w = (lane < nw) ? iscr[lane] : 0;
#pragma unroll
    for (int o = 1; o < 32; o <<= 1) {
      int t = __shfl_up(w, o, 32);
      if (lane >= o) w += t;
    }
    if (lane < nw) iscr[lane] = w;  // inclusive wave-total scan
  }
  __syncthreads();
  int wave_base = (wid == 0) ? 0 : iscr[wid - 1];
  int total = iscr[nw - 1];
  __syncthreads();
  *total_out = total;
  return wave_base + (x - v);
}

// per-thread partial of sum(relu(x - tau)) over the full LDS row
__device__ __forceinline__ float relu_sum_partial(const float* s_row, int d, float tau) {
  const int tid = threadIdx.x;
  const int nv4 = d >> 2;
  const float4* r4 = (const float4*)s_row;
  float acc = 0.0f;
  for (int i = tid; i < nv4; i += BLOCK) {
    float4 v = r4[i];
    acc += fmaxf(v.x - tau, 0.0f);
    acc += fmaxf(v.y - tau, 0.0f);
    acc += fmaxf(v.z - tau, 0.0f);
    acc += fmaxf(v.w - tau, 0.0f);
  }
  for (int i = (nv4 << 2) + tid; i < d; i += BLOCK)
    acc += fmaxf(s_row[i] - tau, 0.0f);
  return acc;
}

__global__ __launch_bounds__(BLOCK, 2)
void sparsemax_bisect_row_kernel(const float* __restrict__ logits,
                                 const int* __restrict__ target,
                                 float* __restrict__ row_loss,
                                 int n, int d, int n_iter) {
  __shared__ __align__(16) float s_row[MAX_D];
  __shared__ __align__(16) float s_cand[CAP];
  __shared__ float s_scr[32];
  __shared__ int   s_iscr[32];

  const int row = blockIdx.x;
  const int tid = threadIdx.x;
  const float* g_row = logits + (size_t)row * (size_t)d;

  // ---- Stage the row into LDS with the Tensor Data Mover (CDNA5 TDM) ----
#if __has_builtin(__builtin_amdgcn_tensor_load_to_lds)
  if (tid < 32) {  // one wave issues the DMA; TDM ignores EXEC
    unsigned lds_off = (unsigned)(unsigned long long)(size_t)(const void*)s_row;
    unsigned long long ga = (unsigned long long)(size_t)(const void*)g_row;
    unsigned ud = (unsigned)d;
    u32x4 g0;                                             // D# group 0
    g0[0] = 1u;                                           // count=1
    g0[1] = lds_off;                                      // lds_addr
    g0[2] = (unsigned)ga;                                 // global_addr lo
    g0[3] = (unsigned)((ga >> 32) & 0x1FFFFFFu) | 0x80000000u;  // addr hi | type=2
    i32x8 g1;                                             // D# group 1
    g1[0] = (int)(2u << 16);                              // data_size = 4 bytes
    g1[1] = (int)((ud & 0xFFFFu) << 16);                  // tensor_dim0 lo16
    g1[2] = (int)(((ud >> 16) & 0xFFFFu) | (1u << 16));   // dim0 hi16 | tensor_dim1=1
    g1[3] = (int)((ud & 0xFFFFu) << 16);                  // tile_dim0 = d
    g1[4] = 1;                                            // tile_dim1=1
    g1[5] = (int)ud;                                      // dim0_stride lo32
    g1[6] = (int)((ud & 0xFFFFu) << 16);                  // stride1 lo16
    g1[7] = (int)(ud >> 16);                              // stride1 hi32
    i32x4 g2 = {0, 0, 0, 0};
    i32x4 g3 = {0, 0, 0, 0};
    i32x8 g4 = {0, 0, 0, 0, 0, 0, 0, 0};
    __builtin_amdgcn_tensor_load_to_lds(g0, g1, g2, g3, g4, 0);
#if __has_builtin(__builtin_amdgcn_s_wait_tensorcnt)
    __builtin_amdgcn_s_wait_tensorcnt(0);
#else
    asm volatile("s_wait_tensorcnt 0x0" ::: "memory");
#endif
  }
  __syncthreads();
#else
  {
    const int nv4 = d >> 2;
    float4* s4 = (float4*)s_row;
    const float4* gg = (const float4*)g_row;
    for (int i = tid; i < nv4; i += BLOCK) s4[i] = gg[i];
    for (int i = (nv4 << 2) + tid; i < d; i += BLOCK) s_row[i] = g_row[i];
    __syncthreads();
  }
#endif

  const int nv4 = d >> 2;
  const float4* r4 = (const float4*)s_row;

  // ---- pass A: row max ----
  float m = -3.402823466e38f;
  for (int i = tid; i < nv4; i += BLOCK) {
    float4 v = r4[i];
    m = fmaxf(m, fmaxf(fmaxf(v.x, v.y), fmaxf(v.z, v.w)));
  }
  for (int i = (nv4 << 2) + tid; i < d; i += BLOCK) m = fmaxf(m, s_row[i]);
  const float max_val = block_reduce_max(m, s_scr);

  const float tau_lo0 = max_val - 1.0f;
  const float dm0 = 1.0f - 1.0f / (float)d;

  // ---- pass B1: count candidates (x > max-1) and f_lo in one sweep ----
  int cnt = 0;
  float fl = 0.0f;
  for (int i = tid; i < nv4; i += BLOCK) {
    float4 v = r4[i];
    float r;
    r = v.x - tau_lo0; if (r > 0.0f) { ++cnt; fl += r; }
    r = v.y - tau_lo0; if (r > 0.0f) { ++cnt; fl += r; }
    r = v.z - tau_lo0; if (r > 0.0f) { ++cnt; fl += r; }
    r = v.w - tau_lo0; if (r > 0.0f) { ++cnt; fl += r; }
  }
  for (int i = (nv4 << 2) + tid; i < d; i += BLOCK) {
    float r = s_row[i] - tau_lo0; if (r > 0.0f) { ++cnt; fl += r; }
  }
  const float f_lo = block_reduce_sum(fl, s_scr) - 1.0f;

  int nc;
  int base = block_exscan_total(cnt, s_iscr, &nc);

  // ---- pass B2: deterministic compaction (same traversal order as B1) ----
  if (nc <= CAP) {
    int w = base;
    for (int i = tid; i < nv4; i += BLOCK) {
      float4 v = r4[i];
      if (v.x > tau_lo0) s_cand[w++] = v.x;
      if (v.y > tau_lo0) s_cand[w++] = v.y;
      if (v.z > tau_lo0) s_cand[w++] = v.z;
      if (v.w > tau_lo0) s_cand[w++] = v.w;
    }
    for (int i = (nv4 << 2) + tid; i < d; i += BLOCK) {
      float v = s_row[i];
      if (v > tau_lo0) s_cand[w++] = v;
    }
  }
  __syncthreads();

  if (nc <= 32 * REG) {
    // ===== fast path: wave 0 does all 50 iterations in registers, no barriers =====
    if (tid < 32) {
      float vals[REG];
#pragma unroll
      for (int j = 0; j < REG; ++j) {
        int idx = tid + 32 * j;
        vals[j] = (idx < nc) ? s_cand[idx] : (tau_lo0 - 1.0f);  // pad contributes 0
      }
      float tau_lo = tau_lo0, dmm = dm0, tau_m = tau_lo0;
      for (int it = 0; it < n_iter; ++it) {
        dmm *= 0.5f;
        tau_m = tau_lo + dmm;
        float a = 0.0f;
#pragma unroll
        for (int j = 0; j < REG; ++j) a += fmaxf(vals[j] - tau_m, 0.0f);
        float f_m = warp_reduce_sum(a) - 1.0f;
        if (f_m * f_lo >= 0.0f) tau_lo = tau_m;
      }
      float a1 = 0.0f, a2 = 0.0f, a3 = 0.0f;
#pragma unroll
      for (int j = 0; j < REG; ++j) {
        float v = vals[j];
        float r = fmaxf(v - tau_m, 0.0f);
        a1 += r; a2 += r * r; a3 += r * v;
      }
      a1 = warp_reduce_sum(a1);
      a2 = warp_reduce_sum(a2);
      a3 = warp_reduce_sum(a3);
      if (tid == 0) {
        float inv = 1.0f / a1;
        float omega = 0.5f * (1.0f - a2 * inv * inv);
        float dot = a3 * inv;
        int t = target[row];
        t = (t < 0) ? 0 : ((t >= d) ? d - 1 : t);
        row_loss[row] = omega + dot - s_row[t];
      }
    }
  } else if (nc <= CAP) {
    // ===== mid path: all threads bisect over compacted list =====
    float tau_lo = tau_lo0, dmm = dm0, tau_m = tau_lo0;
    for (int it = 0; it < n_iter; ++it) {
      dmm *= 0.5f;
      tau_m = tau_lo + dmm;
      float a = 0.0f;
      for (int i = tid; i < nc; i += BLOCK) a += fmaxf(s_cand[i] - tau_m, 0.0f);
      float f_m = block_reduce_sum(a, s_scr) - 1.0f;
      if (f_m * f_lo >= 0.0f) tau_lo = tau_m;  // uniform branch
    }
    float a1 = 0.0f, a2 = 0.0f, a3 = 0.0f;
    for (int i = tid; i < nc; i += BLOCK) {
      float v = s_cand[i];
      float r = fmaxf(v - tau_m, 0.0f);
      a1 += r; a2 += r * r; a3 += r * v;
    }
    float s1 = block_reduce_sum(a1, s_scr);
    float s2 = block_reduce_sum(a2, s_scr);
    float s3 = block_reduce_sum(a3, s_scr);
    if (tid == 0) {
      float inv = 1.0f / s1;
      float omega = 0.5f * (1.0f - s2 * inv * inv);
      float dot = s3 * inv;
      int t = target[row];
      t = (t < 0) ? 0 : ((t >= d) ? d - 1 : t);
      row_loss[row] = omega + dot - s_row[t];
    }
  } else {
    // ===== fallback: full-row sweeps (adversarial inputs) =====
    float tau_lo = tau_lo0, dmm = dm0, tau_m = tau_lo0;
    for (int it = 0; it < n_iter; ++it) {
      dmm *= 0.5f;
      tau_m = tau_lo + dmm;
      float f_m = block_reduce_sum(relu_sum_partial(s_row, d, tau_m), s_scr) - 1.0f;
      if (f_m * f_lo >= 0.0f) tau_lo = tau_m;
    }
    float a1 = 0.0f, a2 = 0.0f, a3 = 0.0f;
    for (int i = tid; i < nv4; i += BLOCK) {
      float4 v = r4[i];
      float r;
      r = fmaxf(v.x - tau_m, 0.0f); a1 += r; a2 += r * r; a3 += r * v.x;
      r = fmaxf(v.y - tau_m, 0.0f); a1 += r; a2 += r * r; a3 += r * v.y;
      r = fmaxf(v.z - tau_m, 0.0f); a1 += r; a2 += r * r; a3 += r * v.z;
      r = fmaxf(v.w - tau_m, 0.0f); a1 += r; a2 += r * r; a3 += r * v.w;
    }
    for (int i = (nv4 << 2) + tid; i < d; i += BLOCK) {
      float v = s_row[i];
      float r = fmaxf(v - tau_m, 0.0f);
      a1 += r; a2 += r * r; a3 += r * v;
    }
    float s1 = block_reduce_sum(a1, s_scr);
    float s2 = block_reduce_sum(a2, s_scr);
    float s3 = block_reduce_sum(a3, s_scr);
    if (tid == 0) {
      float inv = 1.0f / s1;
      float omega = 0.5f * (1.0f - s2 * inv * inv);
      float dot = s3 * inv;
      int t = target[row];
      t = (t < 0) ? 0 : ((t >= d) ? d - 1 : t);
      row_loss[row] = omega + dot - s_row[t];
    }
  }
}

__global__ __launch_bounds__(1024)
void mean_reduce_kernel(const float* __restrict__ row_loss,
                        float* __restrict__ out, int n) {
  __shared__ float scr[32];
  float s = 0.0f;
  for (int i = threadIdx.x; i < n; i += blockDim.x) s += row_loss[i];
  s = block_reduce_sum(s, scr);
  if (threadIdx.x == 0) out[0] = s / (float)n;
}

extern "C" void kernel_launch(void* const* d_in, const int* in_sizes, int n_in,
                              void* d_out, int out_size, void* d_ws, size_t ws_size,
                              hipStream_t stream) {
  const float* logits = (const float*)d_in[0];
  const int* target   = (const int*)d_in[1];
  const int n = in_sizes[1];            // 4096 rows
  const int d = in_sizes[0] / n;        // 32000 classes
  float* row_loss = (float*)d_ws;       // n floats of scratch

  sparsemax_bisect_row_kernel<<<n, BLOCK, 0, stream>>>(logits, target, row_loss,
                                                       n, d, N_ITER);
  mean_reduce_kernel<<<1, 1024, 0, stream>>>(row_loss, (float*)d_out, n);
}